// CompGCNEncoder_68307159875880
// MI455X (gfx1250) — compile-verified
//
#include <hip/hip_runtime.h>
#include <hip/hip_bf16.h>

#define NUM_NODES 100000
#define NUM_RELS  200
#define DIM       128
#define NUM_EDGES 600000

// Padded LDS row stride for B tiles: rows k and k+2 must map to disjoint
// bank sets (64 banks). 144*2 = 288 == 32 (mod 64) -> half-waves disjoint.
#define BSTRIDE   144

typedef float v2f __attribute__((ext_vector_type(2)));
typedef float v8f __attribute__((ext_vector_type(8)));

// ---------------------------------------------------------------------------
// Zero fill (float4 granularity). n4 = number of float4 elements.
// ---------------------------------------------------------------------------
__global__ void zero_kernel(float4* __restrict__ p, int n4) {
    int i = blockIdx.x * blockDim.x + threadIdx.x;
    if (i < n4) p[i] = make_float4(0.f, 0.f, 0.f, 0.f);
}

// ---------------------------------------------------------------------------
// Edge gather + composition + scatter-add.
// One wave (32 lanes) per edge; each lane handles 4 consecutive dims (float4).
// Gather of x[src] / rel[etype] is a fully coalesced 512B burst per wave.
// Scatter uses fp32 atomic add -> GLOBAL_ATOMIC_ADD_F32 resolved in L2
// (agg is 51MB, resident in the 192MB L2).
// ---------------------------------------------------------------------------
__global__ void edge_scatter_kernel(const float* __restrict__ x,
                                    const float* __restrict__ rel,
                                    const int*   __restrict__ src,
                                    const int*   __restrict__ dst,
                                    const int*   __restrict__ etype,
                                    const float* __restrict__ ew,
                                    float* __restrict__ agg,
                                    int num_edges) {
    int wave = (blockIdx.x * blockDim.x + threadIdx.x) >> 5;
    int lane = threadIdx.x & 31;
    if (wave >= num_edges) return;

    int   s = src[wave];
    int   d = dst[wave];
    int   t = etype[wave];
    float w = ew[wave];

    const float4* xs = (const float4*)(x   + (long)s * DIM);
    const float4* rs = (const float4*)(rel + (long)t * DIM);
    float4 xv = xs[lane];
    float4 rv = rs[lane];

    float4 m;
    m.x = (xv.x - rv.x) * w;
    m.y = (xv.y - rv.y) * w;
    m.z = (xv.z - rv.z) * w;
    m.w = (xv.w - rv.w) * w;

    float* a = agg + (long)d * DIM + lane * 4;
    atomicAdd(a + 0, m.x);
    atomicAdd(a + 1, m.y);
    atomicAdd(a + 2, m.z);
    atomicAdd(a + 3, m.w);
}

// ---------------------------------------------------------------------------
// Node GEMM: out[16-row slab] = A0 @ B0 + A1 @ B1   (optionally ReLU)
// A0 = agg [N,128], A1 = x [N,128], B0 = W, B1 = Wself ([128,128] row-major).
//
// Two phases; in each phase the block cooperatively stages one 64KB B matrix
// into LDS (padded stride, bank-conflict-free for the WMMA B read pattern),
// then every wave runs K=128 in steps of 4 with V_WMMA_F32_16X16X4_F32
// (exact fp32). Accumulators persist across phases, fusing the "+".
//
// fp32 WMMA layout (ISA 7.12.2, 32-bit):
//   lane: r = lane&15, kb = (lane>>4)*2
//   A vgpr pair = A[row r][k+kb], A[row r][k+kb+1]
//   B vgpr pair = B[k+kb][col r], B[k+kb+1][col r]
//   C/D vgpr i  = row (i + (lane<16?0:8)), col r
//
// No early returns: all waves reach every __syncthreads() and every WMMA
// executes with EXEC all-1s; tail waves clamp their tile and skip stores.
// ---------------------------------------------------------------------------
__global__ void gemm_node_kernel(const float* __restrict__ A0,
                                 const float* __restrict__ A1,
                                 const float* __restrict__ B0,
                                 const float* __restrict__ B1,
                                 float* __restrict__ out,
                                 int n_row_tiles, int do_relu) {
    __shared__ float Bs[DIM * BSTRIDE];   // 73,728 B of the 320KB WGP LDS

    int wave   = blockIdx.x * (blockDim.x >> 5) + (threadIdx.x >> 5);
    int active = (wave < n_row_tiles);
    int waveC  = active ? wave : (n_row_tiles - 1);

    int lane = threadIdx.x & 31;
    int r    = lane & 15;
    int kb   = (lane >> 4) << 1;           // 0 or 2
    long rowBase = (long)waveC * 16;

    const float* aPtr[2];
    aPtr[0] = A0 + (rowBase + r) * DIM;
    aPtr[1] = A1 + (rowBase + r) * DIM;
    const float* bPtr[2];
    bPtr[0] = B0;
    bPtr[1] = B1;

    v8f acc[8];
#pragma unroll
    for (int t = 0; t < 8; ++t) acc[t] = (v8f)(0.f);

    for (int p = 0; p < 2; ++p) {
        __syncthreads();
        // Cooperative stage: 128x128 floats, float4 per thread per step.
        // 4096 float4s / 256 threads = 16 steps; within a row the 32
        // consecutive threads write 32 consecutive float4s (b128 ds stores).
        for (int idx = threadIdx.x; idx < (DIM * DIM) / 4; idx += blockDim.x) {
            int row = idx >> 5;                 // 32 float4 per row
            int c4  = (idx & 31);
            float4 v = ((const float4*)(bPtr[p] + row * DIM))[c4];
            float* d = &Bs[row * BSTRIDE + c4 * 4];
            d[0] = v.x; d[1] = v.y; d[2] = v.z; d[3] = v.w;
        }
        __syncthreads();

        const float* a = aPtr[p];
        for (int k = 0; k < DIM; k += 4) {
            v2f av;
            av.x = a[k + kb];
            av.y = a[k + kb + 1];
#pragma unroll
            for (int t = 0; t < 8; ++t) {
                int col = t * 16 + r;
                v2f bv;
                bv.x = Bs[(k + kb) * BSTRIDE + col];
                bv.y = Bs[(k + kb + 1) * BSTRIDE + col];
                acc[t] = __builtin_amdgcn_wmma_f32_16x16x4_f32(
                    false, av, false, bv, (short)0, acc[t], false, false);
            }
        }
    }

    if (active) {
        long rr = rowBase + (lane < 16 ? 0 : 8);
#pragma unroll
        for (int t = 0; t < 8; ++t) {
#pragma unroll
            for (int i = 0; i < 8; ++i) {
                float v = acc[t][i];
                if (do_relu) v = fmaxf(v, 0.f);
                out[(rr + i) * DIM + t * 16 + r] = v;
            }
        }
    }
}

// ---------------------------------------------------------------------------
// Rel GEMM: out = rel @ Wrel (optionally ReLU), rows=200 (13 tiles, last partial).
// Tiny problem: direct global B reads (L0-resident). A reads clamp the row
// index (EXEC stays all-1s for WMMA); stores are per-element predicated.
// ---------------------------------------------------------------------------
__global__ void gemm_rel_kernel(const float* __restrict__ A,
                                const float* __restrict__ B,
                                float* __restrict__ out,
                                int rows, int do_relu) {
    int n_row_tiles = (rows + 15) / 16;
    int wave = blockIdx.x * (blockDim.x >> 5) + (threadIdx.x >> 5);
    if (wave >= n_row_tiles) return;
    int lane = threadIdx.x & 31;
    int r    = lane & 15;
    int kb   = (lane >> 4) << 1;
    int rowBase = wave * 16;
    int ar = rowBase + r;
    if (ar > rows - 1) ar = rows - 1;      // clamp A reads, keep EXEC uniform

    const float* a = A + (long)ar * DIM;

    v8f acc[8];
#pragma unroll
    for (int t = 0; t < 8; ++t) acc[t] = (v8f)(0.f);

    for (int k = 0; k < DIM; k += 4) {
        v2f aV;
        aV.x = a[k + kb];  aV.y = a[k + kb + 1];
#pragma unroll
        for (int t = 0; t < 8; ++t) {
            int col = t * 16 + r;
            v2f bV;
            bV.x = B[(k + kb) * DIM + col];
            bV.y = B[(k + kb + 1) * DIM + col];
            acc[t] = __builtin_amdgcn_wmma_f32_16x16x4_f32(
                false, aV, false, bV, (short)0, acc[t], false, false);
        }
    }

    int rbase = rowBase + (lane < 16 ? 0 : 8);
#pragma unroll
    for (int t = 0; t < 8; ++t) {
#pragma unroll
        for (int i = 0; i < 8; ++i) {
            int row = rbase + i;
            if (row < rows) {
                float v = acc[t][i];
                if (do_relu) v = fmaxf(v, 0.f);
                out[(long)row * DIM + t * 16 + r] = v;
            }
        }
    }
}

// ---------------------------------------------------------------------------
// Host-side launch sequence (graph-capture safe: kernels only, all on stream)
// ---------------------------------------------------------------------------
extern "C" void kernel_launch(void* const* d_in, const int* in_sizes, int n_in,
                              void* d_out, int out_size, void* d_ws, size_t ws_size,
                              hipStream_t stream) {
    const int*   edge_index = (const int*)  d_in[0];   // [2, E]
    const int*   etype      = (const int*)  d_in[1];   // [E]
    const float* ew         = (const float*)d_in[2];   // [E]
    const float* node_emb   = (const float*)d_in[3];   // [N, D]
    const float* rel_emb    = (const float*)d_in[4];   // [R, D]
    const float* W0         = (const float*)d_in[5];
    const float* Wself0     = (const float*)d_in[6];
    const float* Wrel0      = (const float*)d_in[7];
    const float* W1         = (const float*)d_in[8];
    const float* Wself1     = (const float*)d_in[9];
    const float* Wrel1      = (const float*)d_in[10];

    const int* src = edge_index;
    const int* dst = edge_index + NUM_EDGES;

    // workspace partition
    float* agg  = (float*)d_ws;                          // N*D
    float* h1   = agg + (size_t)NUM_NODES * DIM;         // N*D
    float* rel1 = h1  + (size_t)NUM_NODES * DIM;         // R*D

    float* outX   = (float*)d_out;                       // [N, D]
    float* outRel = outX + (size_t)NUM_NODES * DIM;      // [R, D]

    const int aggN4       = NUM_NODES * DIM / 4;         // float4 count
    const int zeroBlocks  = (aggN4 + 255) / 256;
    const int edgeBlocks  = (NUM_EDGES * 32 + 255) / 256;
    const int nRowTiles   = NUM_NODES / 16;              // 6250 (exact)
    const int gemmBlocks  = (nRowTiles + 7) / 8;         // 8 waves / block
    const int relTiles    = (NUM_RELS + 15) / 16;        // 13

    // ---------------- layer 0 ----------------
    zero_kernel<<<zeroBlocks, 256, 0, stream>>>((float4*)agg, aggN4);
    edge_scatter_kernel<<<edgeBlocks, 256, 0, stream>>>(
        node_emb, rel_emb, src, dst, etype, ew, agg, NUM_EDGES);
    gemm_node_kernel<<<gemmBlocks, 256, 0, stream>>>(
        agg, node_emb, W0, Wself0, h1, nRowTiles, /*relu=*/1);
    gemm_rel_kernel<<<relTiles, 32, 0, stream>>>(
        rel_emb, Wrel0, rel1, NUM_RELS, /*relu=*/1);

    // ---------------- layer 1 ----------------
    zero_kernel<<<zeroBlocks, 256, 0, stream>>>((float4*)agg, aggN4);
    edge_scatter_kernel<<<edgeBlocks, 256, 0, stream>>>(
        h1, rel1, src, dst, etype, ew, agg, NUM_EDGES);
    gemm_node_kernel<<<gemmBlocks, 256, 0, stream>>>(
        agg, h1, W1, Wself1, outX, nRowTiles, /*relu=*/0);
    gemm_rel_kernel<<<relTiles, 32, 0, stream>>>(
        rel1, Wrel1, outRel, NUM_RELS, /*relu=*/0);
}